// MultiScaleFedGNN_36421322670514
// MI455X (gfx1250) — compile-verified
//
#include <hip/hip_runtime.h>
#include <hip/hip_bf16.h>
#include <math.h>

// Problem dims (from reference)
#define USR_NUM 100000
#define USR_DIM 64
#define NUM_LOC 5000
#define T_STEPS 16
#define N_EDGES 80000
// usr_dim1 = 32 (LSTM0 input), usr_dim2 = 16 (hidden), 4*16 = 64 gate width

typedef float v2f __attribute__((ext_vector_type(2)));
typedef float v8f __attribute__((ext_vector_type(8)));

// ---------------------------------------------------------------- utilities
__global__ void k_zero(float* __restrict__ p, int n) {
  int i = blockIdx.x * blockDim.x + threadIdx.x;
  if (i < n) p[i] = 0.0f;
}

// deg[dst] += 1 over location-graph edges
__global__ void k_deg_accum(const int* __restrict__ dst, float* __restrict__ deg, int E) {
  int i = blockIdx.x * blockDim.x + threadIdx.x;
  if (i < E) atomicAdd(&deg[dst[i]], 1.0f);
}

// deg += 1 (self loop); inv_sqrt = rsqrt(deg); inv_deg = 1/deg
__global__ void k_deg_fin(const float* __restrict__ deg, float* __restrict__ isq,
                          float* __restrict__ idg, int n) {
  int i = blockIdx.x * blockDim.x + threadIdx.x;
  if (i < n) {
    float d = deg[i] + 1.0f;
    isq[i] = rsqrtf(d);
    idg[i] = 1.0f / d;
  }
}

// node2edge scatter: e[ids[i]][d] += (W_emb[i][d] + b_emb[d]); cnt[ids[i]] += 1
__global__ void k_scatter_users(const int* __restrict__ ids, const float* __restrict__ Wemb,
                                const float* __restrict__ bemb, float* __restrict__ e,
                                float* __restrict__ cnt, int N) {
  int t = blockIdx.x * blockDim.x + threadIdx.x;
  if (t >= N * 64) return;
  int i = t >> 6, d = t & 63;
  int id = ids[i];
  atomicAdd(&e[(long)id * 64 + d], Wemb[t] + bemb[d]);
  if (d == 0) atomicAdd(&cnt[id], 1.0f);
}

// e /= max(cnt,1); agg = e * inv_deg   (GCN self-loop init)
__global__ void k_enorm_selfinit(const float* __restrict__ cnt, const float* __restrict__ idg,
                                 float* __restrict__ e, float* __restrict__ agg, int nloc) {
  int t = blockIdx.x * blockDim.x + threadIdx.x;
  if (t >= nloc * 64) return;
  int j = t >> 6;
  float cv = cnt[j]; if (cv < 1.0f) cv = 1.0f;
  float v = e[t] / cv;
  e[t] = v;
  agg[t] = v * idg[j];
}

// agg = h * inv_deg (self-loop init for second GCN layer)
__global__ void k_selfinit(const float* __restrict__ h, const float* __restrict__ idg,
                           float* __restrict__ agg, int nloc) {
  int t = blockIdx.x * blockDim.x + threadIdx.x;
  if (t >= nloc * 64) return;
  agg[t] = h[t] * idg[t >> 6];
}

// agg[dst] += h[src] * inv_sqrt[src] * inv_sqrt[dst]  over location-graph edges
__global__ void k_edge_scatter(const int* __restrict__ src, const int* __restrict__ dst,
                               const float* __restrict__ isq, const float* __restrict__ h,
                               float* __restrict__ agg, int E) {
  int t = blockIdx.x * blockDim.x + threadIdx.x;
  if (t >= E * 64) return;
  int e = t >> 6, d = t & 63;
  int s = src[e], q = dst[e];
  atomicAdd(&agg[(long)q * 64 + d], h[(long)s * 64 + d] * isq[s] * isq[q]);
}

// ------------------------------------------------------------- WMMA GEMMs
// C[M, NT*16] = (relu?) (A[M,K] @ B[K, NT*16] + bias); K <= 64, K % 4 == 0.
// 256-thread block = 8 waves; B staged in LDS once per block (ds_load reuse),
// each wave computes one 16-row tile across the FULL output width (A fragment
// loaded once per K-step, amortized over NT WMMAs).
template <int NT>
__global__ __launch_bounds__(256) void k_gemm_wmma(
    const float* __restrict__ A, const float* __restrict__ B,
    const float* __restrict__ bias, float* __restrict__ C, int M, int K, int relu) {
  constexpr int Nc = NT * 16;
  __shared__ float Bs[64 * Nc];
  for (int i = threadIdx.x; i < K * Nc; i += 256) Bs[i] = B[i];
  __syncthreads();

  const int lane = threadIdx.x & 31;
  const int wave = threadIdx.x >> 5;
  const int half = lane >> 4;    // K pair select (A/B), M+8 (C)
  const int l16  = lane & 15;
  const int row0 = (blockIdx.x * 8 + wave) * 16;

  int ar = row0 + l16; if (ar > M - 1) ar = M - 1;   // clamp tail reads
  const float* Arow = A + (long)ar * K;

  v8f acc[NT];
#pragma unroll
  for (int tt = 0; tt < NT; ++tt) {
    float bv = bias[tt * 16 + l16];
    v8f tmp = { bv, bv, bv, bv, bv, bv, bv, bv };
    acc[tt] = tmp;
  }

  for (int k = 0; k < K; k += 4) {
    v2f a;
    a[0] = Arow[k + 2 * half + 0];
    a[1] = Arow[k + 2 * half + 1];
#pragma unroll
    for (int tt = 0; tt < NT; ++tt) {
      v2f b;
      b[0] = Bs[(k + 2 * half + 0) * Nc + tt * 16 + l16];
      b[1] = Bs[(k + 2 * half + 1) * Nc + tt * 16 + l16];
      acc[tt] = __builtin_amdgcn_wmma_f32_16x16x4_f32(false, a, false, b, (short)0,
                                                      acc[tt], false, false);
    }
  }
#pragma unroll
  for (int tt = 0; tt < NT; ++tt) {
    for (int r = 0; r < 8; ++r) {
      int row = row0 + r + 8 * half;
      if (row < M) {
        float v = acc[tt][r];
        if (relu) v = v > 0.0f ? v : 0.0f;
        C[(long)row * Nc + tt * 16 + l16] = v;
      }
    }
  }
}

// G[M,64] = X[M,Kx] @ Wi[Kx,64] + H[M,16] @ Wh[16,64] + b[64]
// Fused two-phase K loop into one WMMA accumulator; both weight matrices
// staged in LDS (Wi rows [0,Kx), Wh rows [Kx,Kx+16)).
__global__ __launch_bounds__(256) void k_lstm_gates(
    const float* __restrict__ X, int Kx, const float* __restrict__ Wi,
    const float* __restrict__ H, const float* __restrict__ Wh,
    const float* __restrict__ b, float* __restrict__ G, int M) {
  __shared__ float Bs[64 * 64];
  for (int i = threadIdx.x; i < Kx * 64; i += 256) Bs[i] = Wi[i];
  for (int i = threadIdx.x; i < 16 * 64; i += 256) Bs[Kx * 64 + i] = Wh[i];
  __syncthreads();

  const int lane = threadIdx.x & 31;
  const int wave = threadIdx.x >> 5;
  const int half = lane >> 4;
  const int l16  = lane & 15;
  const int row0 = (blockIdx.x * 8 + wave) * 16;

  int ar = row0 + l16; if (ar > M - 1) ar = M - 1;

  v8f acc[4];
#pragma unroll
  for (int tt = 0; tt < 4; ++tt) {
    float bv = b[tt * 16 + l16];
    v8f tmp = { bv, bv, bv, bv, bv, bv, bv, bv };
    acc[tt] = tmp;
  }

  const float* Xr = X + (long)ar * Kx;
  for (int k = 0; k < Kx; k += 4) {
    v2f a;
    a[0] = Xr[k + 2 * half + 0];
    a[1] = Xr[k + 2 * half + 1];
#pragma unroll
    for (int tt = 0; tt < 4; ++tt) {
      v2f w;
      w[0] = Bs[(k + 2 * half + 0) * 64 + tt * 16 + l16];
      w[1] = Bs[(k + 2 * half + 1) * 64 + tt * 16 + l16];
      acc[tt] = __builtin_amdgcn_wmma_f32_16x16x4_f32(false, a, false, w, (short)0,
                                                      acc[tt], false, false);
    }
  }
  const float* Hr = H + (long)ar * 16;
#pragma unroll
  for (int k = 0; k < 16; k += 4) {
    v2f a;
    a[0] = Hr[k + 2 * half + 0];
    a[1] = Hr[k + 2 * half + 1];
#pragma unroll
    for (int tt = 0; tt < 4; ++tt) {
      v2f w;
      w[0] = Bs[(Kx + k + 2 * half + 0) * 64 + tt * 16 + l16];
      w[1] = Bs[(Kx + k + 2 * half + 1) * 64 + tt * 16 + l16];
      acc[tt] = __builtin_amdgcn_wmma_f32_16x16x4_f32(false, a, false, w, (short)0,
                                                      acc[tt], false, false);
    }
  }
#pragma unroll
  for (int tt = 0; tt < 4; ++tt) {
    for (int r = 0; r < 8; ++r) {
      int row = row0 + r + 8 * half;
      if (row < M) G[(long)row * 64 + tt * 16 + l16] = acc[tt][r];
    }
  }
}

// ---------------------------------------------------------- elementwise LSTM
__device__ __forceinline__ float sigm(float x) { return 1.0f / (1.0f + expf(-x)); }

__global__ void k_lstm_cell(const float* __restrict__ G, float* __restrict__ h,
                            float* __restrict__ c, int N) {
  int t = blockIdx.x * blockDim.x + threadIdx.x;
  if (t >= N * 16) return;
  int i = t >> 4, j = t & 15;
  const float* g = G + (long)i * 64;
  float gi = g[j], gf = g[16 + j], gg = g[32 + j], go = g[48 + j];
  float cn = sigm(gf) * c[t] + sigm(gi) * tanhf(gg);
  c[t] = cn;
  h[t] = sigm(go) * tanhf(cn);
}

// out32[i][d] = he2n[ids[i]][d]
__global__ void k_gather_e2n(const int* __restrict__ ids, const float* __restrict__ he,
                             float* __restrict__ out32, int N) {
  int t = blockIdx.x * blockDim.x + threadIdx.x;
  if (t >= N * 32) return;
  int i = t >> 5, d = t & 31;
  out32[t] = he[(long)ids[i] * 32 + d];
}

// out[t][i][c] = h1[i] . W_out[:,c] + b_out[c]
__global__ void k_outproj(const float* __restrict__ h, const float* __restrict__ Wout,
                          const float* __restrict__ bout, float* __restrict__ out, int N) {
  int i = blockIdx.x * blockDim.x + threadIdx.x;
  if (i >= N) return;
  const float* hr = h + (long)i * 16;
  float a0 = bout[0], a1 = bout[1];
  for (int j = 0; j < 16; ++j) {
    float v = hr[j];
    a0 += v * Wout[j * 2 + 0];
    a1 += v * Wout[j * 2 + 1];
  }
  out[(long)i * 2 + 0] = a0;
  out[(long)i * 2 + 1] = a1;
}

// --------------------------------------------------------------------- host
static inline int gs(long n, int b) { return (int)((n + b - 1) / b); }

extern "C" void kernel_launch(void* const* d_in, const int* in_sizes, int n_in,
                              void* d_out, int out_size, void* d_ws, size_t ws_size,
                              hipStream_t stream) {
  (void)in_sizes; (void)n_in; (void)out_size; (void)ws_size;

  const int*   seq  = (const int*)d_in[0];    // [T, N]
  const int*   gidx = (const int*)d_in[1];    // [2, E]
  const float* Wemb = (const float*)d_in[2];  // [N, 64]
  const float* bemb = (const float*)d_in[3];  // [64]
  const float* W1   = (const float*)d_in[4];  // [64, 64]
  const float* b1   = (const float*)d_in[5];
  const float* W2   = (const float*)d_in[6];  // [64, 64]
  const float* b2   = (const float*)d_in[7];
  const float* We2n = (const float*)d_in[8];  // [64, 32]
  const float* be2n = (const float*)d_in[9];
  const float* Wi0  = (const float*)d_in[10]; // [32, 64]
  const float* Wh0  = (const float*)d_in[11]; // [16, 64]
  const float* bL0  = (const float*)d_in[12]; // [64]
  const float* Wi1  = (const float*)d_in[13]; // [16, 64]
  const float* Wh1  = (const float*)d_in[14]; // [16, 64]
  const float* bL1  = (const float*)d_in[15]; // [64]
  const float* Wout = (const float*)d_in[16]; // [16, 2]
  const float* bout = (const float*)d_in[17]; // [2]

  const int* srcE = gidx;
  const int* dstE = gidx + N_EDGES;
  float* out = (float*)d_out;

  // workspace carve (floats); e and cnt kept adjacent for a single zero pass
  float* p = (float*)d_ws;
  float* e     = p; p += (size_t)NUM_LOC * 64;
  float* cnt   = p; p += NUM_LOC;
  float* deg   = p; p += NUM_LOC;
  float* isq   = p; p += NUM_LOC;
  float* idg   = p; p += NUM_LOC;
  float* agg   = p; p += (size_t)NUM_LOC * 64;
  float* h1b   = p; p += (size_t)NUM_LOC * 64;
  float* h2b   = p; p += (size_t)NUM_LOC * 64;
  float* he2n  = p; p += (size_t)NUM_LOC * 32;
  float* out32 = p; p += (size_t)USR_NUM * 32;
  float* gbuf  = p; p += (size_t)USR_NUM * 64;
  float* h0    = p; p += (size_t)USR_NUM * 16;   // h0,c0,h1s,c1s contiguous
  float* c0    = p; p += (size_t)USR_NUM * 16;
  float* h1s   = p; p += (size_t)USR_NUM * 16;
  float* c1s   = p; p += (size_t)USR_NUM * 16;

  const int BT = 256;

  // graph normalization (once per call; deg must be re-zeroed for replay)
  k_zero<<<gs(NUM_LOC, BT), BT, 0, stream>>>(deg, NUM_LOC);
  k_deg_accum<<<gs(N_EDGES, BT), BT, 0, stream>>>(dstE, deg, N_EDGES);
  k_deg_fin<<<gs(NUM_LOC, BT), BT, 0, stream>>>(deg, isq, idg, NUM_LOC);

  // LSTM state init (h0,c0,h1s,c1s contiguous)
  k_zero<<<gs((long)USR_NUM * 64, BT), BT, 0, stream>>>(h0, USR_NUM * 64);

  for (int t = 0; t < T_STEPS; ++t) {
    const int* ids = seq + (long)t * USR_NUM;

    // node2edge mean aggregation
    k_zero<<<gs((long)NUM_LOC * 65, BT), BT, 0, stream>>>(e, NUM_LOC * 65);
    k_scatter_users<<<gs((long)USR_NUM * 64, BT), BT, 0, stream>>>(ids, Wemb, bemb, e, cnt, USR_NUM);
    k_enorm_selfinit<<<gs((long)NUM_LOC * 64, BT), BT, 0, stream>>>(cnt, idg, e, agg, NUM_LOC);

    // GCN layer 1: agg = selfloop + edge scatter; h1 = relu(agg@W1 + b1)
    k_edge_scatter<<<gs((long)N_EDGES * 64, BT), BT, 0, stream>>>(srcE, dstE, isq, e, agg, N_EDGES);
    k_gemm_wmma<4><<<gs(NUM_LOC, 128), 256, 0, stream>>>(agg, W1, b1, h1b, NUM_LOC, 64, 1);

    // GCN layer 2
    k_selfinit<<<gs((long)NUM_LOC * 64, BT), BT, 0, stream>>>(h1b, idg, agg, NUM_LOC);
    k_edge_scatter<<<gs((long)N_EDGES * 64, BT), BT, 0, stream>>>(srcE, dstE, isq, h1b, agg, N_EDGES);
    k_gemm_wmma<4><<<gs(NUM_LOC, 128), 256, 0, stream>>>(agg, W2, b2, h2b, NUM_LOC, 64, 0);

    // edge2node linear, then gather back to users
    k_gemm_wmma<2><<<gs(NUM_LOC, 128), 256, 0, stream>>>(h2b, We2n, be2n, he2n, NUM_LOC, 64, 0);
    k_gather_e2n<<<gs((long)USR_NUM * 32, BT), BT, 0, stream>>>(ids, he2n, out32, USR_NUM);

    // LSTM layer 0 step
    k_lstm_gates<<<gs(USR_NUM, 128), 256, 0, stream>>>(out32, 32, Wi0, h0, Wh0, bL0, gbuf, USR_NUM);
    k_lstm_cell<<<gs((long)USR_NUM * 16, BT), BT, 0, stream>>>(gbuf, h0, c0, USR_NUM);

    // LSTM layer 1 step (input = fresh h0)
    k_lstm_gates<<<gs(USR_NUM, 128), 256, 0, stream>>>(h0, 16, Wi1, h1s, Wh1, bL1, gbuf, USR_NUM);
    k_lstm_cell<<<gs((long)USR_NUM * 16, BT), BT, 0, stream>>>(gbuf, h1s, c1s, USR_NUM);

    // output projection for this timestep
    k_outproj<<<gs(USR_NUM, BT), BT, 0, stream>>>(h1s, Wout, bout, out + (long)t * USR_NUM * 2, USR_NUM);
  }
}